// VanillaDiameterExtractor_73675868996059
// MI455X (gfx1250) — compile-verified
//
#include <hip/hip_runtime.h>
#include <stdint.h>

#define NR   24     // radii per group
#define NZ   512    // z per (b,f)
#define NBF  128    // nb*nf = 16*8
#define ZC   256    // z chunk per workgroup (2 per thread)
#define NTHR 128    // threads per block (4 wave32)

typedef uint32_t u32x4 __attribute__((ext_vector_type(4)));
typedef int      i32x4 __attribute__((ext_vector_type(4)));
typedef int      i32x8 __attribute__((ext_vector_type(8)));
typedef float    f32x2 __attribute__((ext_vector_type(2)));

// M2C[k] = -2*cos(k * 15 deg), k = j - i.   d2_ij = s_i + s_j + M2C[j-i]*(r_i*r_j)
// constexpr so every use folds to an inline constant after full unroll.
constexpr float M2C[NR] = {
  -2.0f,        -1.93185165f, -1.73205081f, -1.41421356f, -1.0f, -0.51763809f,
   0.0f,         0.51763809f,  1.0f,         1.41421356f,  1.73205081f, 1.93185165f,
   2.0f,         1.93185165f,  1.73205081f,  1.41421356f,  1.0f,  0.51763809f,
   0.0f,        -0.51763809f, -1.0f,        -1.41421356f, -1.73205081f, -1.93185165f
};

__device__ static inline f32x2 splat2(float v) { f32x2 r; r.x = v; r.y = v; return r; }

// Squared distance between points i and j for two z columns (packed f32).
// i, j are compile-time constants after unrolling, so the branches fold.
__device__ static inline f32x2 pair_d2(const f32x2 r[NR], const f32x2 s[NR],
                                       int i, int j) {
  const int k = j - i;
  if (k == 12) {                       // cos = -1: d2 = (r_i + r_j)^2
    f32x2 w = r[i] + r[j];
    return w * w;                      // pk_add + pk_mul
  }
  f32x2 u = s[i] + s[j];               // pk_add
  if (k == 6 || k == 18) return u;     // cos = 0: no cross term
  f32x2 t = r[i] * r[j];               // pk_mul
  return __builtin_elementwise_fma(splat2(M2C[k]), t, u);   // pk_fma
}

// One 2-D TDM descriptor: tile_w contiguous fp32 elements x tile_h rows,
// rows strided by row_stride_elems in global memory, packed densely into LDS.
__device__ static inline void tdm_load_tile_2d(uint32_t lds_off, const void* gaddr,
                                               uint32_t tile_w, uint32_t tile_h,
                                               uint32_t row_stride_elems,
                                               uint32_t tensor_w, uint32_t tensor_h) {
  uint64_t ga = (uint64_t)(uintptr_t)gaddr;
  u32x4 g0;
  g0.x = 1u;                                     // count=1 (valid user descriptor)
  g0.y = lds_off;                                // lds_addr (bytes)
  g0.z = (uint32_t)ga;                           // global_addr[31:0]
  g0.w = ((uint32_t)(ga >> 32) & 0x01FFFFFFu)    // global_addr[56:32]
         | (2u << 30);                           // type = 2 ("image")
  i32x8 g1;
  g1[0] = (int)(2u << 16);                       // workgroup_mask=0, data_size=2 (4B)
  g1[1] = (int)((tensor_w & 0xFFFFu) << 16);     // tensor_dim0[15:0]
  g1[2] = (int)(((tensor_w >> 16) & 0xFFFFu) |
                ((tensor_h & 0xFFFFu) << 16));   // tensor_dim0[31:16] | tensor_dim1[15:0]
  g1[3] = (int)(((tensor_h >> 16) & 0xFFFFu) |
                ((tile_w & 0xFFFFu) << 16));     // tensor_dim1[31:16] | tile_dim0
  g1[4] = (int)(tile_h & 0xFFFFu);               // tile_dim1 ; tile_dim2=0
  g1[5] = (int)row_stride_elems;                 // tensor_dim0_stride[31:0]
  g1[6] = 0;                                     // stride[47:32]=0 ; dim1_stride lo=0
  g1[7] = 0;                                     // dim1_stride hi=0
  i32x4 gz = {0, 0, 0, 0};
#if defined(__clang_major__) && (__clang_major__ >= 23)
  i32x8 gz8 = {0, 0, 0, 0, 0, 0, 0, 0};
  __builtin_amdgcn_tensor_load_to_lds(g0, g1, gz, gz, gz8, 0);
#else
  __builtin_amdgcn_tensor_load_to_lds(g0, g1, gz, gz, 0);
#endif
}

__global__ __launch_bounds__(NTHR)
void diameter_extract_kernel(const float* __restrict__ x, float* __restrict__ out) {
  __shared__ float tile[NR * ZC];                // 24 x 256 fp32 = 24 KB

  const int bf  = blockIdx.y;                    // 0..127  (b*8+f)
  const int zc  = blockIdx.x;                    // 0..1    (z chunk)
  const int tid = threadIdx.x;                   // 0..127; handles z = 2*tid, 2*tid+1

  const float* gtile = x + (size_t)bf * (NR * NZ) + (size_t)zc * ZC;

  // Wave 0 issues the TDM DMA for the whole workgroup's tile.
  if (tid < 32) {
    uint32_t lds_off = (uint32_t)(uintptr_t)&tile[0];  // low 32b of LDS aperture addr
    tdm_load_tile_2d(lds_off, gtile,
                     /*tile_w=*/ZC, /*tile_h=*/NR,
                     /*row_stride=*/NZ,
                     /*tensor_w=*/NZ, /*tensor_h=*/NR);
    __builtin_amdgcn_s_wait_tensorcnt((short)0);
  }
  __syncthreads();

  f32x2 r[NR], s[NR];
#pragma unroll
  for (int i = 0; i < NR; ++i) {
    f32x2 v = *(const f32x2*)&tile[i * ZC + 2 * tid];  // ds_load_b64, conflict-free
    r[i] = v;
    s[i] = v * v;                                      // r_i^2 (pk_mul)
  }

  // 4 rotating accumulators break the max dependency chain; two distances per
  // step so the maxnum chain can fuse into v_max3_num_f32.
  f32x2 acc[4] = { splat2(0.0f), splat2(0.0f), splat2(0.0f), splat2(0.0f) };
#pragma unroll
  for (int i = 0; i < NR; ++i) {
    f32x2 m = acc[i & 3];
    int j = i + 1;
#pragma unroll
    for (; j + 1 < NR; j += 2) {
      f32x2 d1 = pair_d2(r, s, i, j);
      f32x2 d2 = pair_d2(r, s, i, j + 1);
      m = __builtin_elementwise_max(__builtin_elementwise_max(m, d1), d2);
    }
#pragma unroll
    for (; j < NR; ++j) {
      m = __builtin_elementwise_max(m, pair_d2(r, s, i, j));
    }
    acc[i & 3] = m;
  }
  f32x2 m01 = __builtin_elementwise_max(acc[0], acc[1]);
  f32x2 m23 = __builtin_elementwise_max(acc[2], acc[3]);
  f32x2 m   = __builtin_elementwise_max(m01, m23);

  f32x2 res;
  res.x = __builtin_sqrtf(m.x);
  res.y = __builtin_sqrtf(m.y);
  *(f32x2*)&out[(size_t)bf * NZ + (size_t)zc * ZC + 2 * tid] = res;  // b64 store
}

extern "C" void kernel_launch(void* const* d_in, const int* in_sizes, int n_in,
                              void* d_out, int out_size, void* d_ws, size_t ws_size,
                              hipStream_t stream) {
  (void)in_sizes; (void)n_in; (void)out_size; (void)d_ws; (void)ws_size;
  const float* x = (const float*)d_in[0];
  float* out = (float*)d_out;
  dim3 grid(NZ / ZC, NBF);   // (2, 128) = 256 workgroups
  dim3 block(NTHR);          // 128 threads = 4 wave32
  diameter_extract_kernel<<<grid, block, 0, stream>>>(x, out);
}